// ShootingModel_62783831933357
// MI455X (gfx1250) — compile-verified
//
#include <hip/hip_runtime.h>

// ---------------------------------------------------------------------------
// ShootingModel on MI455X (gfx1250, wave32).
//   x_{t+1} = x_t @ W + bw  (N=4096, T=1024);  y = x @ R + br (K=64)
//
// Strategy: block-16 shooting.
//   W16 = W^16 via 4 WMMA-f32 squarings (exact fp32 dtype, RNE).
//   Serial prefix rows 1..16 via deterministic split-K GEMV.
//   d = x16 - x0 @ W16   (affine constant of the 16-step map).
//   rows[16j..16j+15] = rows[16(j-1)..] @ W16 + d : 63 WMMA GEMMs (M=16).
//   Readout y = x @ R + br : 1 WMMA GEMM.
// ---------------------------------------------------------------------------

#define NDIM 4096
#define KOUT 64
#define TSTEPS 1024
#define BLK 16                    // time-block size (W^16)
#define GEMV_CHUNKS 32            // split-K chunks (4096/128)
#define GEMV_CHUNK_I 128

typedef float v2f __attribute__((ext_vector_type(2)));
typedef float v8f __attribute__((ext_vector_type(8)));
typedef int   v4i __attribute__((vector_size(4 * sizeof(int))));
typedef __attribute__((address_space(1))) v4i* gbl_v4i_p;   // global ptr
typedef __attribute__((address_space(3))) v4i* lds_v4i_p;   // LDS ptr

#if defined(__HIP_DEVICE_COMPILE__) && __has_builtin(__builtin_amdgcn_global_load_async_to_lds_b128)
#define HAVE_ASYNC_LDS 1
#else
#define HAVE_ASYNC_LDS 0
#endif

__device__ __forceinline__ void wait_async_zero()
{
#if __has_builtin(__builtin_amdgcn_s_wait_asynccnt)
  __builtin_amdgcn_s_wait_asynccnt(0);
#else
  asm volatile("s_wait_asynccnt 0" ::: "memory");
#endif
}

// Small deterministic scratch in device globals (no host symbol lookups).
__device__ float g_partial[GEMV_CHUNKS * NDIM];   // split-K GEMV partials
__device__ float g_tmp[NDIM];                      // x0 @ W16
__device__ float g_d[NDIM];                        // 16-step affine constant

// ---------------------------------------------------------------------------
// Split-K GEMV pass 1: g_partial[c][j] = sum_{i in chunk c} x[i] * W[i][j]
// grid = (NDIM/256, GEMV_CHUNKS), block = 256
// ---------------------------------------------------------------------------
__global__ __launch_bounds__(256) void gemv_partial_k(
    const float* __restrict__ x, const float* __restrict__ W)
{
  __shared__ float xs[GEMV_CHUNK_I];
  const int j  = blockIdx.x * 256 + threadIdx.x;
  const int c  = blockIdx.y;
  const int i0 = c * GEMV_CHUNK_I;
  for (int u = threadIdx.x; u < GEMV_CHUNK_I; u += 256) xs[u] = x[i0 + u];
  __syncthreads();
  const float* Wp = W + (size_t)i0 * NDIM + j;
  float acc = 0.f;
#pragma unroll 8
  for (int i = 0; i < GEMV_CHUNK_I; ++i)
    acc = fmaf(xs[i], Wp[(size_t)i * NDIM], acc);
  g_partial[c * NDIM + j] = acc;
}

// ---------------------------------------------------------------------------
// Split-K GEMV pass 2: out[j] = bias[j] + sum_c partial[c][j]
// to_tmp!=0 writes g_tmp instead of out. grid = NDIM/256, block = 256
// ---------------------------------------------------------------------------
__global__ __launch_bounds__(256) void gemv_reduce_k(
    const float* __restrict__ bias, float* __restrict__ out, int to_tmp)
{
  const int j = blockIdx.x * 256 + threadIdx.x;
  float s = bias ? bias[j] : 0.f;
#pragma unroll
  for (int c = 0; c < GEMV_CHUNKS; ++c) s += g_partial[c * NDIM + j];
  if (to_tmp) g_tmp[j] = s; else out[j] = s;
}

// d[j] = x16[j] - (x0 @ W16)[j]
__global__ __launch_bounds__(256) void make_d_k(const float* __restrict__ X)
{
  const int j = blockIdx.x * 256 + threadIdx.x;
  g_d[j] = X[(size_t)BLK * NDIM + j] - g_tmp[j];
}

// ---------------------------------------------------------------------------
// FP32 WMMA GEMM: C[M,N] = A[M,K] @ B[K,N] + bias[N]   (row-major)
// V_WMMA_F32_16X16X4_F32, wave32. Block tile 128x128, 8 waves, each wave
// owns a 32x64 subtile = 2x4 accumulators of 16x16.
//
// LDS layouts (operand = single conflict-free b64 load per lane):
//   As[m][k], row stride KPP=20 floats  -> a = (A[m][k], A[m][k+1])
//   Bs pair-interleaved: float2 (B[2p][n], B[2p+1][n]), pair-row stride 144
// A tile is staged via GLOBAL_LOAD_ASYNC_TO_LDS_B128 (ASYNCcnt) when the
// toolchain exposes the builtin; synchronous float4 staging otherwise.
// bias_mode: 0 = none, 1 = use `bias` pointer, 2 = use g_d.
// ---------------------------------------------------------------------------
#define BM 128
#define BN 128
#define KP 16
#define KPP 20      // A LDS row stride (floats)
#define BNP2 144    // B LDS pair-row stride (float2 units); 2*144 % 64 == 32

__global__ __launch_bounds__(256) void wmma_gemm_f32_k(
    const float* __restrict__ A, const float* __restrict__ B,
    float* __restrict__ C, const float* __restrict__ bias, int bias_mode,
    int M, int N, int K, int lda, int ldb, int ldc)
{
  __shared__ float As[BM * KPP];             // 10240 B
  __shared__ float Bs[(KP / 2) * BNP2 * 2];  // 9216 B

  const int tid  = threadIdx.x;
  const int lane = tid & 31;
  const int wave = tid >> 5;      // 0..7
  const int wm   = wave & 3;      // 4 waves along M
  const int wn   = wave >> 2;     // 2 waves along N
  const int m0   = blockIdx.y * BM;
  const int n0   = blockIdx.x * BN;

  v8f acc[2][4];
#pragma unroll
  for (int i = 0; i < 2; ++i)
#pragma unroll
    for (int j = 0; j < 4; ++j)
#pragma unroll
      for (int e = 0; e < 8; ++e) acc[i][j][e] = 0.f;

  const int lm    = lane & 15;
  const int khalf = (lane < 16) ? 0 : 2;   // K split across lane halves
  const int sp    = tid >> 5;              // B staging: pair row 0..7
  const int snq   = (tid & 31) << 2;       // B staging: col 0..124

  for (int kp = 0; kp < K; kp += KP) {
    // ---- stage A tile (BM x KP) row-major into As[m][k] ----
#pragma unroll
    for (int f = tid; f < (BM * KP) / 4; f += 256) {
      const int r = f >> 2;            // row in tile
      const int q = (f & 3) << 2;      // k sub-offset 0,4,8,12
      int gm = m0 + r;
      if (gm >= M) gm = 0;             // safe redirect; rows >= M never stored
      const float* gp = A + (size_t)gm * lda + kp + q;
#if HAVE_ASYNC_LDS
      __builtin_amdgcn_global_load_async_to_lds_b128(
          (gbl_v4i_p)(void*)gp, (lds_v4i_p)(void*)&As[r * KPP + q], 0, 0);
#else
      *(float4*)(&As[r * KPP + q]) = *(const float4*)gp;
#endif
    }
    // ---- stage B tile (KP x BN) pair-interleaved: one (2p,2p+1) row pair ----
    {
      float4 v0 = make_float4(0.f, 0.f, 0.f, 0.f);
      float4 v1 = v0;
      const int gk = kp + 2 * sp;
      const int gn = n0 + snq;
      if (gn + 3 < N) {                // N is always a multiple of 4 here
        v0 = *(const float4*)(B + (size_t)gk * ldb + gn);
        v1 = *(const float4*)(B + (size_t)(gk + 1) * ldb + gn);
      }
      v2f* bsr = (v2f*)&Bs[(sp * BNP2 + snq) * 2];
      bsr[0] = (v2f){v0.x, v1.x};
      bsr[1] = (v2f){v0.y, v1.y};
      bsr[2] = (v2f){v0.z, v1.z};
      bsr[3] = (v2f){v0.w, v1.w};
    }
#if HAVE_ASYNC_LDS
    wait_async_zero();
#endif
    __syncthreads();

    // ---- 4 k-steps of 4, 8 WMMAs each; operands are direct b64 loads ----
#pragma unroll
    for (int kk = 0; kk < KP; kk += 4) {
      v2f a[2], b[4];
#pragma unroll
      for (int i = 0; i < 2; ++i) {
        const int m = 32 * wm + 16 * i + lm;
        a[i] = *(const v2f*)(&As[m * KPP + kk + khalf]);
      }
      const int pr = (kk >> 1) + ((lane < 16) ? 0 : 1);
#pragma unroll
      for (int j = 0; j < 4; ++j) {
        const int n = 64 * wn + 16 * j + lm;
        b[j] = *(const v2f*)(&Bs[(pr * BNP2 + n) * 2]);
      }
#pragma unroll
      for (int i = 0; i < 2; ++i)
#pragma unroll
        for (int j = 0; j < 4; ++j)
          acc[i][j] = __builtin_amdgcn_wmma_f32_16x16x4_f32(
              false, a[i], false, b[j], (short)0, acc[i][j], false, false);
    }
    __syncthreads();
  }

  // ---- store: lane(0-15)=N col, VGPR r = M row (+8 for upper lanes) ----
  const float* bp = (bias_mode == 2) ? g_d : (bias_mode == 1 ? bias : nullptr);
  const int mh = (lane < 16) ? 0 : 8;
#pragma unroll
  for (int j = 0; j < 4; ++j) {
    const int col = n0 + 64 * wn + 16 * j + lm;
    if (col >= N) continue;
    const float bv = bp ? bp[col] : 0.f;
#pragma unroll
    for (int i = 0; i < 2; ++i) {
#pragma unroll
      for (int r = 0; r < 8; ++r) {
        const int row = m0 + 32 * wm + 16 * i + mh + r;
        if (row < M) C[(size_t)row * ldc + col] = acc[i][j][r] + bv;
      }
    }
  }
}

// ---------------------------------------------------------------------------
// Host side
// ---------------------------------------------------------------------------
static inline void launch_gemm(const float* A, const float* B, float* C,
                               const float* bias, int bias_mode,
                               int M, int N, int K, int lda, int ldb, int ldc,
                               hipStream_t stream)
{
  dim3 grid((N + BN - 1) / BN, (M + BM - 1) / BM);
  wmma_gemm_f32_k<<<grid, 256, 0, stream>>>(A, B, C, bias, bias_mode,
                                            M, N, K, lda, ldb, ldc);
}

static inline void launch_gemv(const float* x, const float* W,
                               const float* bias, float* out, int to_tmp,
                               hipStream_t stream)
{
  dim3 g1(NDIM / 256, GEMV_CHUNKS);
  gemv_partial_k<<<g1, 256, 0, stream>>>(x, W);
  gemv_reduce_k<<<NDIM / 256, 256, 0, stream>>>(bias, out, to_tmp);
}

extern "C" void kernel_launch(void* const* d_in, const int* in_sizes, int n_in,
                              void* d_out, int out_size, void* d_ws, size_t ws_size,
                              hipStream_t stream)
{
  (void)in_sizes; (void)n_in; (void)out_size;

  const float* x0 = (const float*)d_in[0];
  const float* W  = (const float*)d_in[1];
  const float* bw = (const float*)d_in[2];
  const float* R  = (const float*)d_in[3];
  const float* br = (const float*)d_in[4];
  // d_in[5] = T (device scalar); T/N/K are module constants, hardcoded.

  float* X = (float*)d_out;                              // [1024][4096]
  float* Y = (float*)d_out + (size_t)TSTEPS * NDIM;      // [1024][64]

  // row 0 of the trajectory is x0
  (void)hipMemcpyAsync(X, x0, NDIM * sizeof(float), hipMemcpyDeviceToDevice, stream);

  const size_t SZ_W = (size_t)NDIM * NDIM * sizeof(float);   // 64 MB

  if (ws_size >= 2 * SZ_W) {
    // ---------------- fast path: block-16 shooting ----------------
    float* buf0 = (float*)d_ws;
    float* buf1 = (float*)((char*)d_ws + SZ_W);

    // W16 = W^16 by repeated squaring (fp32 WMMA, RNE)
    launch_gemm(W,    W,    buf0, nullptr, 0, NDIM, NDIM, NDIM, NDIM, NDIM, NDIM, stream); // W^2
    launch_gemm(buf0, buf0, buf1, nullptr, 0, NDIM, NDIM, NDIM, NDIM, NDIM, NDIM, stream); // W^4
    launch_gemm(buf1, buf1, buf0, nullptr, 0, NDIM, NDIM, NDIM, NDIM, NDIM, NDIM, stream); // W^8
    launch_gemm(buf0, buf0, buf1, nullptr, 0, NDIM, NDIM, NDIM, NDIM, NDIM, NDIM, stream); // W^16
    const float* W16 = buf1;

    // serial prefix: rows 1..16
    for (int t = 0; t < BLK; ++t)
      launch_gemv(X + (size_t)t * NDIM, W, bw, X + (size_t)(t + 1) * NDIM, 0, stream);

    // d = x16 - x0 @ W16
    launch_gemv(X, W16, nullptr, nullptr, /*to_tmp=*/1, stream);
    make_d_k<<<NDIM / 256, 256, 0, stream>>>(X);

    // advance 16 rows at a time: rows[16j..] = rows[16(j-1)..] @ W16 + d
    for (int j = 1; j < TSTEPS / BLK; ++j)
      launch_gemm(X + (size_t)(j - 1) * BLK * NDIM, W16,
                  X + (size_t)j * BLK * NDIM,
                  nullptr, /*bias_mode=g_d*/ 2,
                  BLK, NDIM, NDIM, NDIM, NDIM, NDIM, stream);
  } else {
    // ---------------- fallback: plain serial recurrence ----------------
    for (int t = 0; t < TSTEPS - 1; ++t)
      launch_gemv(X + (size_t)t * NDIM, W, bw, X + (size_t)(t + 1) * NDIM, 0, stream);
  }

  // readout: y = x @ R + br   (M=1024, N=64, K=4096)
  launch_gemm(X, R, Y, br, 1, TSTEPS, KOUT, NDIM, NDIM, KOUT, KOUT, stream);
}